// NEPNN_90726889160848
// MI455X (gfx1250) — compile-verified
//
#include <hip/hip_runtime.h>

#define NE 262144
#define NT 16     // pair types
#define NH 32     // DES_HIDDEN
#define NB 16     // BASIS
#define NN 8      // N_MAX
#define BN_EPS 1e-5f

typedef __attribute__((ext_vector_type(2))) float v2f;
typedef __attribute__((ext_vector_type(8))) float v8f;

__device__ __forceinline__ float silu_f(float x) {
    return x / (1.0f + __expf(-x));
}

// D = A(16x4 f32) * B(4x16 f32) + C(16x16 f32), exact f32 reference semantics.
__device__ __forceinline__ v8f wmma_k4(v2f a, v2f b, v8f c) {
    return __builtin_amdgcn_wmma_f32_16x16x4_f32(
        /*neg_a=*/false, a, /*neg_b=*/false, b,
        /*c_mod=*/(short)0, c, /*reuse_a=*/false, /*reuse_b=*/false);
}

// -------- zero the stats slots in workspace --------
__global__ void k_zero(float* stats) {
    if (threadIdx.x < 80) stats[threadIdx.x] = 0.f;
}

// -------- layer 1: out1[e,0:32] = W0[t_e] @ fn[e] via type-masked WMMA,
// -------- with fused per-channel sum/sumsq partials (BN1 statistics) --------
__global__ __launch_bounds__(256) void k_layer1(const int* __restrict__ bij,
                                                const float* __restrict__ fn,
                                                const float* __restrict__ W0,
                                                float* __restrict__ out1,
                                                float* __restrict__ stats1) {
    __shared__ float sW0[NT * NH * NB];                 // 32 KB; reused as reduce scratch
    for (int i = threadIdx.x; i < NT * NH * NB; i += 256) sW0[i] = W0[i];
    __syncthreads();

    const int lane = threadIdx.x & 31;
    const int col  = lane & 15;                         // pair within tile (B/C column)
    const int half = lane >> 4;                         // K-pair selector for A/B frags
    const int wid  = (blockIdx.x << 3) + (threadIdx.x >> 5);
    const int nw   = gridDim.x << 3;
    const int ntiles = NE / 16;

    // per-thread BN partials: s[v]/q[v] = channels 8*half+v, s[8+v]/q[8+v] = 16+8*half+v
    float s[16], q[16];
    #pragma unroll
    for (int v = 0; v < 16; ++v) { s[v] = 0.f; q[v] = 0.f; }

    for (int tile = wid; tile < ntiles; tile += nw) {
        const int base = tile << 4;
        const int tcol = bij[base + col];

        v2f bv[4];
        #pragma unroll
        for (int kk = 0; kk < 4; ++kk)                  // fn[col][kk*4 + 2*half + {0,1}]
            bv[kk] = *(const v2f*)(fn + (base + col) * NB + kk * 4 + 2 * half);

        v8f acc0 = {0,0,0,0,0,0,0,0};                   // rows h = 0..15
        v8f acc1 = {0,0,0,0,0,0,0,0};                   // rows h = 16..31
        const int m = col;                              // A-matrix row = lane&15

        for (int t = 0; t < NT; ++t) {
            if (__ballot(tcol == t) == 0ull) continue;  // uniform skip: type absent
            #pragma unroll
            for (int kk = 0; kk < 4; ++kk) {
                v2f z = {0.f, 0.f};
                v2f bm = (tcol == t) ? bv[kk] : z;      // mask B columns per type
                v2f a0 = *(const v2f*)(&sW0[t*(NH*NB) + m*NB       + kk*4 + 2*half]);
                v2f a1 = *(const v2f*)(&sW0[t*(NH*NB) + (16+m)*NB  + kk*4 + 2*half]);
                acc0 = wmma_k4(a0, bm, acc0);
                acc1 = wmma_k4(a1, bm, acc1);
            }
        }
        // C layout: lane holds 8 contiguous channels -> 32B vector stores
        *(v8f*)(out1 + (base + col) * NH + 8 * half)      = acc0;
        *(v8f*)(out1 + (base + col) * NH + 16 + 8 * half) = acc1;
        #pragma unroll
        for (int v = 0; v < 8; ++v) {
            s[v]     += acc0[v];  q[v]     += acc0[v] * acc0[v];
            s[8 + v] += acc1[v];  q[8 + v] += acc1[v] * acc1[v];
        }
    }

    // block-reduce partials through LDS (sW0 no longer needed)
    __syncthreads();
    #pragma unroll
    for (int v = 0; v < 16; ++v) {
        sW0[v * 256 + threadIdx.x]        = s[v];       // slots 0..15: sums
        sW0[(16 + v) * 256 + threadIdx.x] = q[v];       // slots 16..31: sumsq
    }
    __syncthreads();
    if (threadIdx.x < 64) {
        const int kind = threadIdx.x >> 5;              // 0 = sum, 1 = sumsq
        const int c    = threadIdx.x & 31;              // channel
        const int ch_half = (c >> 3) & 1;               // which lane-half produced c
        const int slot = (c < 16) ? (c & 7) : (8 + (c & 7));
        const int j = slot + 16 * kind;
        float acc = 0.f;
        for (int w = 0; w < 8; ++w)
            for (int l = 0; l < 16; ++l)
                acc += sW0[j * 256 + w * 32 + ch_half * 16 + l];
        atomicAdd(&stats1[kind * 32 + c], acc);
    }
}

// -------- layer 2: BN1 + SiLU fused into B frags, then W1[t_e] @ y via WMMA,
// -------- with fused per-channel sum/sumsq partials (BN2 statistics) --------
__global__ __launch_bounds__(256) void k_layer2(const int* __restrict__ bij,
                                                const float* __restrict__ out1,
                                                const float* __restrict__ W1,
                                                const float* __restrict__ g1,
                                                const float* __restrict__ b1,
                                                const float* __restrict__ stats1,
                                                const int* __restrict__ normp,
                                                float* __restrict__ out2,
                                                float* __restrict__ stats2) {
    __shared__ float sW1[NT * 16 * NH];                 // rows 8..15 zero-padded, 32 KB
    __shared__ float sMean[NH], sRstd[NH], sG[NH], sB[NH];
    for (int i = threadIdx.x; i < NT * 16 * NH; i += 256) {
        int t = i >> 9; int m = (i >> 5) & 15; int h = i & 31;
        sW1[i] = (m < NN) ? W1[t * (NN * NH) + m * NH + h] : 0.f;
    }
    if (threadIdx.x < NH) {
        float su = stats1[threadIdx.x], ssq = stats1[NH + threadIdx.x];
        float mean = su * (1.f / NE);
        float var  = ssq * (1.f / NE) - mean * mean;    // biased, as torch BN
        sMean[threadIdx.x] = mean;
        sRstd[threadIdx.x] = rsqrtf(var + BN_EPS);
        sG[threadIdx.x] = g1[threadIdx.x];
        sB[threadIdx.x] = b1[threadIdx.x];
    }
    __syncthreads();

    const int nrm  = normp[0];
    const int lane = threadIdx.x & 31;
    const int col  = lane & 15;
    const int half = lane >> 4;
    const int wid  = (blockIdx.x << 3) + (threadIdx.x >> 5);
    const int nw   = gridDim.x << 3;
    const int ntiles = NE / 16;

    float s[8], q[8];                                   // BN2 partials (pad rows == 0)
    #pragma unroll
    for (int v = 0; v < 8; ++v) { s[v] = 0.f; q[v] = 0.f; }

    for (int tile = wid; tile < ntiles; tile += nw) {
        const int base = tile << 4;
        const int tcol = bij[base + col];

        v2f bv[8];                                      // K = 32 -> 8 k-steps
        #pragma unroll
        for (int kk = 0; kk < 8; ++kk) {
            const int c0 = kk * 4 + 2 * half;
            v2f x = *(const v2f*)(out1 + (base + col) * NH + c0);
            float x0 = x.x, x1 = x.y;
            if (nrm) {
                x0 = (x0 - sMean[c0])     * sRstd[c0]     * sG[c0]     + sB[c0];
                x1 = (x1 - sMean[c0 + 1]) * sRstd[c0 + 1] * sG[c0 + 1] + sB[c0 + 1];
            }
            v2f y; y.x = silu_f(x0); y.y = silu_f(x1);
            bv[kk] = y;
        }

        v8f acc = {0,0,0,0,0,0,0,0};
        const int m = col;
        for (int t = 0; t < NT; ++t) {
            if (__ballot(tcol == t) == 0ull) continue;
            #pragma unroll
            for (int kk = 0; kk < 8; ++kk) {
                v2f z = {0.f, 0.f};
                v2f bm = (tcol == t) ? bv[kk] : z;
                v2f a  = *(const v2f*)(&sW1[t * 512 + m * 32 + kk * 4 + 2 * half]);
                acc = wmma_k4(a, bm, acc);
            }
        }
        if (half == 0)                                  // rows 0..7 real; 8..15 pad
            *(v8f*)(out2 + (base + col) * NN) = acc;
        #pragma unroll
        for (int v = 0; v < 8; ++v) {                   // half==1 lanes hold exact zeros
            s[v] += acc[v];  q[v] += acc[v] * acc[v];
        }
    }

    // block-reduce partials through LDS (sW1 no longer needed)
    __syncthreads();
    #pragma unroll
    for (int v = 0; v < 8; ++v) {
        sW1[v * 256 + threadIdx.x]       = s[v];        // slots 0..7: sums
        sW1[(8 + v) * 256 + threadIdx.x] = q[v];        // slots 8..15: sumsq
    }
    __syncthreads();
    if (threadIdx.x < 16) {
        const int kind = threadIdx.x >> 3;              // 0 = sum, 1 = sumsq
        const int c    = threadIdx.x & 7;
        const int j = c + 8 * kind;
        float acc = 0.f;
        for (int t2 = 0; t2 < 256; ++t2) acc += sW1[j * 256 + t2];
        atomicAdd(&stats2[kind * 8 + c], acc);
    }
}

// -------- BN2 + SiLU -> output [E, 8] f32 --------
__global__ __launch_bounds__(256) void k_out(const float* __restrict__ out2,
                                             const float* __restrict__ stats2,
                                             const float* __restrict__ g2,
                                             const float* __restrict__ b2,
                                             const int* __restrict__ normp,
                                             float* __restrict__ out) {
    const int nrm = normp[0];
    int i = blockIdx.x * 256 + threadIdx.x;
    const int stride = gridDim.x * 256;
    for (; i < NE * NN; i += stride) {
        const int c = i & 7;
        float x = out2[i];
        if (nrm) {
            float su = stats2[c], ssq = stats2[8 + c];
            float mean = su * (1.f / NE);
            float var  = ssq * (1.f / NE) - mean * mean;
            x = (x - mean) * rsqrtf(var + BN_EPS) * g2[c] + b2[c];
        }
        out[i] = silu_f(x);
    }
}

extern "C" void kernel_launch(void* const* d_in, const int* in_sizes, int n_in,
                              void* d_out, int out_size, void* d_ws, size_t ws_size,
                              hipStream_t stream) {
    (void)in_sizes; (void)n_in; (void)out_size; (void)ws_size;
    const int*   bij = (const int*)  d_in[0];
    const float* fn  = (const float*)d_in[1];
    const float* W0  = (const float*)d_in[2];
    const float* W1  = (const float*)d_in[3];
    const float* g1  = (const float*)d_in[4];
    const float* b1  = (const float*)d_in[5];
    const float* g2  = (const float*)d_in[6];
    const float* b2  = (const float*)d_in[7];
    const int*   nrm = (const int*)  d_in[8];
    float* out = (float*)d_out;

    float* ws    = (float*)d_ws;
    float* out1  = ws;                                   // E*32 floats (32 MB)
    float* out2  = ws + (size_t)NE * NH;                 // E*8 floats  (8 MB)
    float* stats = out2 + (size_t)NE * NN;               // 80 floats
    float* stats1 = stats;                               // [0..63]  sum/sumsq x32
    float* stats2 = stats + 64;                          // [64..79] sum/sumsq x8

    hipLaunchKernelGGL(k_zero,   dim3(1),   dim3(128), 0, stream, stats);
    hipLaunchKernelGGL(k_layer1, dim3(512), dim3(256), 0, stream,
                       bij, fn, W0, out1, stats1);
    hipLaunchKernelGGL(k_layer2, dim3(512), dim3(256), 0, stream,
                       bij, out1, W1, g1, b1, stats1, nrm, out2, stats2);
    hipLaunchKernelGGL(k_out,    dim3(512), dim3(256), 0, stream,
                       out2, stats2, g2, b2, nrm, out);
}